// MertForNERwithESD_89945205113364
// MI455X (gfx1250) — compile-verified
//
#include <hip/hip_runtime.h>
#include <hip/hip_bf16.h>

typedef float v2f __attribute__((ext_vector_type(2)));
typedef float v8f __attribute__((ext_vector_type(8)));

// Problem constants (from setup_inputs)
#define BB   64
#define LL   512
#define DD   768
#define TT   9
#define TE   3
#define ROWS (BB * LL)          // 32768
#define WSTRIDE 770             // LDS column stride (770 % 64 == 2 -> conflict-free v2f)

// ---------------------------------------------------------------------------
// Fused dual-head GEMM: logits = HS@Wc + bc + (EHS@We + be)@W_e2n
// One wave per 16-row tile, f32 WMMA 16x16x4 along K=768.
// ---------------------------------------------------------------------------
__global__ __launch_bounds__(256) void head_gemm_kernel(
    const float* __restrict__ hs,    // [ROWS, 768]
    const float* __restrict__ ehs,   // [ROWS, 768]
    const float* __restrict__ Wc,    // [768, 9]
    const float* __restrict__ bc,    // [9]
    const float* __restrict__ We,    // [768, 3]
    const float* __restrict__ be,    // [3]
    const float* __restrict__ W_e2n, // [3, 9]
    float* __restrict__ logits_out,  // [ROWS, 9]
    float* __restrict__ esd_out)     // [ROWS, 3]
{
    // Transposed, zero-padded weights in LDS: col n at sW*[n*WSTRIDE + k]
    __shared__ __align__(16) float sWcT[10 * WSTRIDE]; // col 9 = zeros
    __shared__ __align__(16) float sWeT[4  * WSTRIDE]; // col 3 = zeros

    const int tid = threadIdx.x;
    for (int idx = tid; idx < 10 * DD; idx += 256) {
        int n = idx / DD, k = idx % DD;
        sWcT[n * WSTRIDE + k] = (n < TT) ? Wc[k * TT + n] : 0.0f;
    }
    for (int idx = tid; idx < 4 * DD; idx += 256) {
        int n = idx / DD, k = idx % DD;
        sWeT[n * WSTRIDE + k] = (n < TE) ? We[k * TE + n] : 0.0f;
    }
    __syncthreads();

    const int wave = tid >> 5;
    const int lane = tid & 31;
    const int tile = blockIdx.x * 8 + wave;      // 256 blocks * 8 waves = 2048 tiles
    const int row0 = tile * 16;

    const int hf = lane >> 4;                    // half-wave: K offset pair selector
    const int n  = lane & 15;                    // N column (also A-frag M index)
    const int kofs = 2 * hf;

    const float* arow = hs  + (size_t)(row0 + n) * DD;
    const float* erow = ehs + (size_t)(row0 + n) * DD;
    const int nc = (n < TT) ? n : TT;            // clamp to zero-padded column
    const int ne = (n < TE) ? n : TE;
    const float* bcol1 = sWcT + nc * WSTRIDE;
    const float* bcol2 = sWeT + ne * WSTRIDE;

    v8f c1 = {};
    v8f c2 = {};

    for (int k = 0; k < DD; k += 4) {
        // A frag (16x4 f32): lane<16 -> {HS[M][k],HS[M][k+1]}, lane>=16 -> {k+2,k+3}
        v2f a1 = *(const v2f*)(arow + k + kofs);
        v2f a2 = *(const v2f*)(erow + k + kofs);
        // B frag (4x16 f32), same K-pair mapping, column n from LDS
        v2f b1 = *(const v2f*)(bcol1 + k + kofs);
        v2f b2 = *(const v2f*)(bcol2 + k + kofs);
        c1 = __builtin_amdgcn_wmma_f32_16x16x4_f32(false, a1, false, b1,
                                                   (short)0, c1, false, false);
        c2 = __builtin_amdgcn_wmma_f32_16x16x4_f32(false, a2, false, b2,
                                                   (short)0, c2, false, false);
    }

    // Epilogue: biases + ESD->NER fusion (C2[M][0..2] live on lanes 16*hf + e)
    const float bcn = (n < TT) ? bc[n] : 0.0f;
    const float be0 = be[0], be1 = be[1], be2 = be[2];
    const float w0 = (n < TT) ? W_e2n[0 * TT + n] : 0.0f;
    const float w1 = (n < TT) ? W_e2n[1 * TT + n] : 0.0f;
    const float w2 = (n < TT) ? W_e2n[2 * TT + n] : 0.0f;
    const int srcbase = 16 * hf;

#pragma unroll
    for (int r = 0; r < 8; r++) {
        float c2v = c2[r];
        float e0 = __shfl(c2v, srcbase + 0, 32) + be0;
        float e1 = __shfl(c2v, srcbase + 1, 32) + be1;
        float e2 = __shfl(c2v, srcbase + 2, 32) + be2;
        int mrow = row0 + r + 8 * hf;
        if (n < TT)
            logits_out[(size_t)mrow * TT + n] = c1[r] + bcn + e0 * w0 + e1 * w1 + e2 * w2;
        if (n < TE)
            esd_out[(size_t)mrow * TE + n] = c2v + ((n == 0) ? be0 : (n == 1) ? be1 : be2);
    }
}

// ---------------------------------------------------------------------------
// CRF log-likelihood (sum over batch of num - den). One wave per sequence.
// ---------------------------------------------------------------------------
template <int T>
__global__ __launch_bounds__(32) void crf_kernel(
    const float* __restrict__ em,    // [B, L, T]
    const int* __restrict__ tags,    // [B, L]
    const int* __restrict__ mask,    // [B, L]
    const float* __restrict__ start, // [T]
    const float* __restrict__ trans, // [T, T]
    const float* __restrict__ endv,  // [T]
    float* __restrict__ accum)
{
    const int b = blockIdx.x;
    const int lane = threadIdx.x;
    const float* e = em + (size_t)b * LL * T;
    const int* tg = tags + b * LL;
    const int* mk = mask + b * LL;
    const float NEG = -1e30f;

    float trans_col[T];
#pragma unroll
    for (int i = 0; i < T; i++)
        trans_col[i] = (lane < T) ? trans[i * T + lane] : 0.0f;

    float alpha = (lane < T) ? (start[lane] + e[lane]) : NEG;

    int prev = tg[0];
    float num = start[prev] + e[prev];            // uniform on all lanes

    for (int l = 1; l < LL; l++) {
        float el = (lane < T) ? e[l * T + lane] : NEG;
        float m = (float)mk[l];

        // denominator step: nxt_j = LSE_i(alpha_i + trans[i][j]) + e_j
        float ai[T];
        float mx = NEG;
#pragma unroll
        for (int i = 0; i < T; i++) {
            ai[i] = __shfl(alpha, i, 32) + trans_col[i];
            mx = fmaxf(mx, ai[i]);
        }
        float s = 0.0f;
#pragma unroll
        for (int i = 0; i < T; i++)
            s += __expf(ai[i] - mx);
        float nxt = mx + __logf(s) + el;
        if (lane < T) alpha = (m > 0.0f) ? nxt : alpha;

        // numerator step (computed uniformly; scalar loads of trans)
        int t = tg[l];
        float et = __shfl(el, t, 32);
        float tr = trans[prev * T + t];
        num += (tr + et) * m;
        prev = (m > 0.0f) ? t : prev;
    }
    num += endv[prev];

    // den = LSE_j(alpha_j + end_j) via wave reduction
    float v = (lane < T) ? (alpha + endv[lane]) : NEG;
    float mx = v;
#pragma unroll
    for (int off = 16; off > 0; off >>= 1)
        mx = fmaxf(mx, __shfl_xor(mx, off, 32));
    float s = (lane < T) ? __expf(v - mx) : 0.0f;
#pragma unroll
    for (int off = 16; off > 0; off >>= 1)
        s += __shfl_xor(s, off, 32);
    float den = mx + __logf(s);

    if (lane == 0)
        atomicAdd(accum, num - den);
}

__global__ void finalize_kernel(const float* __restrict__ acc,
                                float* __restrict__ out_scalar)
{
    if (threadIdx.x == 0)
        out_scalar[0] = -acc[0] - 0.5f * acc[1];   // loss + RATIO * ESD_loss
}

// ---------------------------------------------------------------------------
extern "C" void kernel_launch(void* const* d_in, const int* in_sizes, int n_in,
                              void* d_out, int out_size, void* d_ws, size_t ws_size,
                              hipStream_t stream)
{
    const float* hs      = (const float*)d_in[0];
    const float* ehs     = (const float*)d_in[1];
    const float* W_cls   = (const float*)d_in[2];
    const float* b_cls   = (const float*)d_in[3];
    const float* W_esd   = (const float*)d_in[4];
    const float* b_esd   = (const float*)d_in[5];
    const float* W_e2n   = (const float*)d_in[6];
    const float* crf_s   = (const float*)d_in[7];
    const float* crf_e   = (const float*)d_in[8];
    const float* crf_t   = (const float*)d_in[9];
    const float* esd_s   = (const float*)d_in[10];
    const float* esd_e   = (const float*)d_in[11];
    const float* esd_t   = (const float*)d_in[12];
    const int*   labels  = (const int*)d_in[13];
    const int*   esd_lab = (const int*)d_in[14];
    const int*   attn    = (const int*)d_in[15];

    float* logits_out = (float*)d_out;             // [ROWS * 9]
    float* loss_out   = (float*)d_out + ROWS * TT; // scalar

    float* acc        = (float*)d_ws;              // acc[0]=NER llh, acc[1]=ESD llh
    float* esd_logits = (float*)d_ws + 16;         // [ROWS * 3]

    hipMemsetAsync(d_ws, 0, 64, stream);

    head_gemm_kernel<<<ROWS / (16 * 8), 256, 0, stream>>>(
        hs, ehs, W_cls, b_cls, W_esd, b_esd, W_e2n, logits_out, esd_logits);

    crf_kernel<TT><<<BB, 32, 0, stream>>>(
        logits_out, labels, attn, crf_s, crf_t, crf_e, acc + 0);

    crf_kernel<TE><<<BB, 32, 0, stream>>>(
        esd_logits, esd_lab, attn, esd_s, esd_t, esd_e, acc + 1);

    finalize_kernel<<<1, 32, 0, stream>>>(acc, loss_out);
}